// ReorderObjectsLayer_10806137717520
// MI455X (gfx1250) — compile-verified
//
#include <hip/hip_runtime.h>

// ---------------------------------------------------------------------------
// ReorderObjects (per-event stream compaction) for MI455X / gfx1250.
//
// x: (N, 16, 3) f32.  Groups: jets [0,6), electrons [6,9), muons [9,12),
// photons [12,15), met [15,16).  Each group: stable-compact objects with
// feature0 > 0 to the front, zero-pad the rest.  met copied verbatim.
//
// Bandwidth-bound: 768 MB total traffic -> ~33 us floor @ 23.3 TB/s.
// Strategy: async-DMA a 256-row tile (48 KB) into LDS with coalesced b128
// transactions (GVS mode: SGPR base + 32-bit VGPR offset), compact each row
// in-registers (branchless select network), async-DMA the tile back out.
// LDS rows padded to 208 B so the 48-dword row stride doesn't alias banks.
// Steady-state blocks take a guard-free path (uniform branch); only the one
// partial tail block pays per-lane EXEC guards.
// ---------------------------------------------------------------------------

#define ROWS_PER_BLOCK 256
#define ROW_F          48    // floats per row
#define ROW_B          192   // bytes per row
#define F4_PER_ROW     12    // float4s per row
#define PAD_ROW_B      208   // padded LDS row stride (13 x 16B; gcd(52,64)=4)

// --- CDNA5 async global<->LDS ops (ASYNCcnt), inline asm (portable across
// --- ROCm 7.2 and amdgpu-toolchain).  GVS addressing: saddr base + v offset.
__device__ __forceinline__ void async_load_b128(unsigned lds_addr, unsigned voff,
                                                const void* sbase) {
    asm volatile("global_load_async_to_lds_b128 %0, %1, %2"
                 :: "v"(lds_addr), "v"(voff), "s"(sbase) : "memory");
}
__device__ __forceinline__ void async_store_b128(unsigned voff, const void* sbase,
                                                 unsigned lds_addr) {
    asm volatile("global_store_async_from_lds_b128 %0, %1, %2"
                 :: "v"(voff), "v"(lds_addr), "s"(sbase) : "memory");
}
__device__ __forceinline__ void wait_async0() {
    asm volatile("s_wait_asynccnt 0" ::: "memory");
}

// Branchless stable compaction of K objects (3 floats each) in place.
// Output slot j receives the unique input i (i >= j) with keep[i] and
// prefix_kept[i] == j; unfilled slots are zeroed.  Fully unrolled,
// constant indices -> registers only.
template<int K>
__device__ __forceinline__ void compact_inplace(float* p) {
    float v[3 * K];
#pragma unroll
    for (int i = 0; i < 3 * K; ++i) v[i] = p[i];

    bool keep[K];
    int  pfx[K];
    int  c = 0;
#pragma unroll
    for (int i = 0; i < K; ++i) {
        keep[i] = v[3 * i] > 0.0f;   // NaN > 0 is false, matches jnp
        pfx[i]  = c;
        c += keep[i] ? 1 : 0;
    }

#pragma unroll
    for (int j = 0; j < K; ++j) {
        float a = 0.0f, b = 0.0f, d = 0.0f;
#pragma unroll
        for (int i = j; i < K; ++i) {   // pfx[i] <= i, so source i >= slot j
            bool sel = keep[i] && (pfx[i] == j);
            a = sel ? v[3 * i + 0] : a;
            b = sel ? v[3 * i + 1] : b;
            d = sel ? v[3 * i + 2] : d;
        }
        p[3 * j + 0] = a;
        p[3 * j + 1] = b;
        p[3 * j + 2] = d;
    }
}

__global__ __launch_bounds__(ROWS_PER_BLOCK)
void reorder_objects_kernel(const float* __restrict__ x,
                            float* __restrict__ out,
                            int n_rows)
{
    __shared__ __align__(16) unsigned char tile[ROWS_PER_BLOCK * PAD_ROW_B];

    const int t        = threadIdx.x;
    const int base_row = (int)blockIdx.x * ROWS_PER_BLOCK;

    const char* gin  = (const char*)x   + (long long)base_row * ROW_B;
    char*       gout = (char*)out       + (long long)base_row * ROW_B;
    const unsigned lds_base = (unsigned)(size_t)(&tile[0]);

    const bool full = (base_row + ROWS_PER_BLOCK) <= n_rows;   // wave-uniform

    // ---- Phase 1: coalesced async fill, global -> LDS (padded rows) -------
    if (full) {
#pragma unroll
        for (int i = 0; i < F4_PER_ROW; ++i) {
            int q = i * ROWS_PER_BLOCK + t;      // float4 id within the tile
            int r = q / F4_PER_ROW;
            int c = q - r * F4_PER_ROW;
            async_load_b128(lds_base + (unsigned)(r * PAD_ROW_B + c * 16),
                            (unsigned)(q * 16), gin);
        }
    } else {
#pragma unroll
        for (int i = 0; i < F4_PER_ROW; ++i) {
            int q = i * ROWS_PER_BLOCK + t;
            int r = q / F4_PER_ROW;
            int c = q - r * F4_PER_ROW;
            if (base_row + r < n_rows) {
                async_load_b128(lds_base + (unsigned)(r * PAD_ROW_B + c * 16),
                                (unsigned)(q * 16), gin);
            }
        }
    }
    wait_async0();
    __syncthreads();

    // ---- Phase 2: each thread compacts its own row in place ---------------
    if (full || (base_row + t) < n_rows) {
        float* row = (float*)(tile + t * PAD_ROW_B);
        compact_inplace<6>(row + 0);    // jets      floats [0,18)
        compact_inplace<3>(row + 18);   // electrons floats [18,27)
        compact_inplace<3>(row + 27);   // muons     floats [27,36)
        compact_inplace<3>(row + 36);   // photons   floats [36,45)
        // met floats [45,48): identity copy -> already in place
    }
    __syncthreads();   // all ds stores visible before the async drain

    // ---- Phase 3: coalesced async drain, LDS -> global --------------------
    if (full) {
#pragma unroll
        for (int i = 0; i < F4_PER_ROW; ++i) {
            int q = i * ROWS_PER_BLOCK + t;
            int r = q / F4_PER_ROW;
            int c = q - r * F4_PER_ROW;
            async_store_b128((unsigned)(q * 16), gout,
                             lds_base + (unsigned)(r * PAD_ROW_B + c * 16));
        }
    } else {
#pragma unroll
        for (int i = 0; i < F4_PER_ROW; ++i) {
            int q = i * ROWS_PER_BLOCK + t;
            int r = q / F4_PER_ROW;
            int c = q - r * F4_PER_ROW;
            if (base_row + r < n_rows) {
                async_store_b128((unsigned)(q * 16), gout,
                                 lds_base + (unsigned)(r * PAD_ROW_B + c * 16));
            }
        }
    }
    // S_ENDPGM performs an implicit wait-idle (all counters, incl. ASYNCcnt).
}

extern "C" void kernel_launch(void* const* d_in, const int* in_sizes, int n_in,
                              void* d_out, int out_size, void* d_ws, size_t ws_size,
                              hipStream_t stream)
{
    (void)n_in; (void)out_size; (void)d_ws; (void)ws_size;
    const float* x   = (const float*)d_in[0];
    float*       out = (float*)d_out;

    const int n_rows = in_sizes[0] / ROW_F;              // (N,16,3) -> N
    const int blocks = (n_rows + ROWS_PER_BLOCK - 1) / ROWS_PER_BLOCK;

    reorder_objects_kernel<<<blocks, ROWS_PER_BLOCK, 0, stream>>>(x, out, n_rows);
}